// MVPCL_10788957847983
// MI455X (gfx1250) — compile-verified
//
#include <hip/hip_runtime.h>
#include <math.h>

#define BATCH 32
#define NTOK  4096
#define DIM   768
#define MAXNC 20
#define PADK  112   // 7 token tiles of 16
#define PADNC 32    // 2 cluster tiles of 16
#define CP    772   // center LDS pitch (stride-4 banks -> conflict-free frags)
#define SUMP  772   // sums LDS pitch
#define SP    36    // score LDS pitch
#define KITERS 10

typedef float v2f __attribute__((ext_vector_type(2)));
typedef float v8f __attribute__((ext_vector_type(8)));

// ---------------- Kernel 1: softmax prob + stable top-k indices ----------------
__global__ __launch_bounds__(256) void topk_kernel(const float* __restrict__ amap,
                                                   const int* __restrict__ pid,
                                                   int* __restrict__ topIdx) {
    __shared__ float probs[NTOK];
    __shared__ float redV[256];
    __shared__ int   redI[256];
    __shared__ int   selIdx[PADK];
    const int b = blockIdx.x, tid = threadIdx.x;
    int nc = MAXNC - 2 * pid[0];
    int ksel = nc * 5; if (ksel > NTOK) ksel = NTOK;
    const float* am = amap + (size_t)b * NTOK * 2;
    for (int i = tid; i < NTOK; i += 256) {
        float a0 = am[2 * i], a1 = am[2 * i + 1];
        probs[i] = 1.0f / (1.0f + expf(a0 - a1));   // softmax(...)[ ,1]
    }
    __syncthreads();
    for (int r = 0; r < ksel; ++r) {
        float bv = -INFINITY; int bi = 0x7fffffff;
        for (int i = tid; i < NTOK; i += 256) {
            float v = probs[i];
            if (v > bv) { bv = v; bi = i; }          // keeps lowest index on ties
        }
        redV[tid] = bv; redI[tid] = bi;
        __syncthreads();
        for (int s = 128; s > 0; s >>= 1) {
            if (tid < s) {
                float v2 = redV[tid + s]; int i2 = redI[tid + s];
                if (v2 > redV[tid] || (v2 == redV[tid] && i2 < redI[tid])) {
                    redV[tid] = v2; redI[tid] = i2;
                }
            }
            __syncthreads();
        }
        if (tid == 0) { selIdx[r] = redI[0]; probs[redI[0]] = -INFINITY; }
        __syncthreads();
    }
    for (int r = tid; r < PADK; r += 256)
        topIdx[b * 128 + r] = (r < ksel) ? selIdx[r] : selIdx[0]; // pad rows -> valid token
}

// ---------------- Kernel 2: fused k-means (WMMA distances) + normalize ----------------
__global__ __launch_bounds__(256) void kmeans_kernel(const float* __restrict__ patch,
                                                     const int* __restrict__ pid,
                                                     const int* __restrict__ topIdx,
                                                     float* __restrict__ out) {
    extern __shared__ float smem[];
    float* cent   = smem;                     // PADNC x CP
    float* sums   = cent + PADNC * CP;        // MAXNC x SUMP
    float* score  = sums + MAXNC * SUMP;      // PADK x SP  (score[token][cluster])
    float* cnorm  = score + PADK * SP;        // PADNC
    int*   idxL   = (int*)(cnorm + PADNC);    // PADK
    int*   assignL = idxL + PADK;             // PADK
    int*   counts  = assignL + PADK;          // PADNC

    const int b = blockIdx.x, tid = threadIdx.x;
    const int lane = tid & 31, wave = tid >> 5;
    int nc = MAXNC - 2 * pid[0];
    int ksel = nc * 5; if (ksel > NTOK) ksel = NTOK;
    const int ttiles = (ksel + 15) >> 4;
    const int ctiles = (nc + 15) >> 4;
    const int njobs = ttiles * ctiles;
    const float* pb = patch + (size_t)b * NTOK * DIM;

    for (int r = tid; r < PADK; r += 256) idxL[r] = topIdx[b * 128 + r];
    __syncthreads();
    // init centers = first nc selected tokens; pad rows zero
    for (int t = tid; t < PADNC * DIM; t += 256) {
        int j = t / DIM, k = t - j * DIM;
        cent[j * CP + k] = (j < nc) ? pb[(size_t)idxL[j] * DIM + k] : 0.0f;
    }
    __syncthreads();

    for (int it = 0; it <= KITERS; ++it) {
        // ||c_j||^2 (pad rows are zero -> cnorm 0, harmless)
        if (tid < PADNC) {
            float s = 0.f;
            for (int k = 0; k < DIM; ++k) { float c = cent[tid * CP + k]; s += c * c; }
            cnorm[tid] = s;
        }
        __syncthreads();

        // distance scores via f32 WMMA: D[m=cluster][n=token] = dot(c_m, t_n)
        for (int job = wave; job < njobs; job += 8) {
            int tm = job % ctiles, tn = job / ctiles;
            int nl = lane & 15;
            int khi = (lane >> 4) << 1;                  // lanes 16..31 carry K+2,K+3
            const float* ap = cent + (tm * 16 + nl) * CP;
            const float* tokp = pb + (size_t)idxL[tn * 16 + nl] * DIM;
            v8f acc = {0.f, 0.f, 0.f, 0.f, 0.f, 0.f, 0.f, 0.f};
#pragma unroll 4
            for (int k0 = 0; k0 < DIM; k0 += 4) {
                v2f a  = *(const v2f*)(ap + k0 + khi);    // A: 16x4 cluster frag (LDS)
                v2f bb = *(const v2f*)(tokp + k0 + khi);  // B: 4x16 token frag (global/L2)
                acc = __builtin_amdgcn_wmma_f32_16x16x4_f32(
                        false, a, false, bb, (short)0, acc, false, false);
            }
            int mo = (lane >> 4) * 8;
            int n = tn * 16 + nl;
#pragma unroll
            for (int v = 0; v < 8; ++v) {
                int m = tm * 16 + mo + v;
                score[n * SP + m] = cnorm[m] - 2.0f * acc[v]; // ||t||^2 const per row: drop
            }
        }
        __syncthreads();

        // argmin over real clusters (strict < == first-min, matches jnp.argmin)
        if (tid < ksel) {
            float best = INFINITY; int bj = 0;
            for (int j = 0; j < nc; ++j) {
                float s = score[tid * SP + j];
                if (s < best) { best = s; bj = j; }
            }
            assignL[tid] = bj;
        }
        for (int t = tid; t < MAXNC * SUMP; t += 256) sums[t] = 0.f;
        if (tid < PADNC) counts[tid] = 0;
        __syncthreads();

        if (tid < ksel) atomicAdd(&counts[assignL[tid]], 1);
        for (int t = tid; t < ksel * DIM; t += 256) {       // ds_add_f32, conflict-free
            int i = t / DIM, d = t - i * DIM;
            atomicAdd(&sums[assignL[i] * SUMP + d], pb[(size_t)idxL[i] * DIM + d]);
        }
        __syncthreads();

        if (it < KITERS) {  // in-loop update: empty clusters keep old centers
            for (int t = tid; t < nc * DIM; t += 256) {
                int j = t / DIM, d = t - j * DIM;
                int cn = counts[j];
                if (cn > 0) cent[j * CP + d] = sums[j * SUMP + d] / (float)cn;
            }
            __syncthreads();
        }
    }

    // final means: empty clusters -> 0 (divisor max(counts,1))
    for (int t = tid; t < nc * DIM; t += 256) {
        int j = t / DIM, d = t - j * DIM;
        int cn = counts[j];
        sums[j * SUMP + d] = (cn > 0) ? sums[j * SUMP + d] / (float)cn : 0.f;
    }
    __syncthreads();
    if (tid < nc) {
        float s = 0.f;
        for (int d = 0; d < DIM; ++d) { float m = sums[tid * SUMP + d]; s += m * m; }
        cnorm[tid] = sqrtf(s);
    }
    __syncthreads();
    for (int t = tid; t < nc * DIM; t += 256) {
        int j = t / DIM, d = t - j * DIM;
        out[(size_t)b * nc * DIM + t] = sums[j * SUMP + d] / fmaxf(cnorm[j], 1e-12f);
    }
}

extern "C" void kernel_launch(void* const* d_in, const int* in_sizes, int n_in,
                              void* d_out, int out_size, void* d_ws, size_t ws_size,
                              hipStream_t stream) {
    (void)in_sizes; (void)n_in; (void)out_size; (void)ws_size;
    const float* patch = (const float*)d_in[0];   // (32,4096,768) f32
    const float* amap  = (const float*)d_in[1];   // (32,4096,2)   f32
    const int*   pid   = (const int*)d_in[2];     // scalar int
    float* out = (float*)d_out;
    int* topIdx = (int*)d_ws;                     // 32 x 128 ints

    topk_kernel<<<BATCH, 256, 0, stream>>>(amap, pid, topIdx);

    size_t smem = (size_t)(PADNC * CP + MAXNC * SUMP + PADK * SP + PADNC) * sizeof(float)
                + (size_t)(PADK + PADK + PADNC) * sizeof(int);   // ~174 KB of 320 KB LDS
    kmeans_kernel<<<BATCH, 256, smem, stream>>>(patch, pid, topIdx, out);
}